// BilinearSamplerLayer_81329500717368
// MI455X (gfx1250) — compile-verified
//
#include <hip/hip_runtime.h>
#include <hip/hip_bf16.h>

// Bilinear sampler for MI455X (gfx1250, wave32).
//
// Bandwidth-bound gather: ~276 MB min HBM traffic -> ~12 us roofline @ 23.3 TB/s.
// WMMA is intentionally NOT used: the 4-tap weighted sum has a per-pixel "B"
// operand (gathered neighbor rows), so it cannot be expressed as a shared-B
// 16x16xK matmul without wasting 15/16 of the MACs; the op is 100x below the
// matrix-core roofline crossover anyway.
//
// MI455X-specific choices:
//  - Non-temporal stores for the 134 MB write-once output and NT loads for the
//    streamed flow (gfx1250 TH cache-policy), so the 134 MB image stays
//    resident in the 192 MB global L2 across its ~4x-reuse gather pattern.
//  - wave32 mapping: 8 lanes per pixel, float4 (b128) per lane. One wave = 4
//    pixels: gathers are 4x128B contiguous runs, stores are 512B contiguous.

typedef float  v4f __attribute__((ext_vector_type(4)));
typedef float  v2f __attribute__((ext_vector_type(2)));

#define B_  16
#define H_  256
#define W_  256
#define C_  32
#define HW_ (H_ * W_)

__global__ __launch_bounds__(256) void bilinear_sampler_kernel(
    const float* __restrict__ image,
    const float* __restrict__ flow,
    float*       __restrict__ out)
{
    // gtid in [0, B*H*W*8): 8 lanes (channel quads) per pixel.
    const int gtid = blockIdx.x * blockDim.x + threadIdx.x;
    const int pix  = gtid >> 3;   // pixel index in [0, B*H*W)
    const int q    = gtid & 7;    // which float4 of the 32 channels

    const int b  = pix >> 16;           // / (H*W) == /65536
    const int hw = pix & (HW_ - 1);
    const int yi = hw >> 8;             // / W
    const int xi = hw & (W_ - 1);

    // Flow: 8B per pixel, read-once stream -> non-temporal load.
    // All 8 lanes of a pixel hit the same address (coalesces to one request).
    const v2f* fp = (const v2f*)flow;
    v2f f = __builtin_nontemporal_load(&fp[pix]);

    // Base grid: linspace(-1,1,N) with step = 2/(N-1), matching jnp.linspace.
    const float stepx = 2.0f / (float)(W_ - 1);
    const float stepy = 2.0f / (float)(H_ - 1);
    const float xb = -1.0f + (float)xi * stepx;
    const float yb = -1.0f + (float)yi * stepy;

    float x = 0.3f * f.x + xb;
    float y = 0.3f * f.y + yb;
    // Reference uses W/2 and H/2 (not (W-1)/2).
    x = (x + 1.0f) * ((float)W_ * 0.5f);
    y = (y + 1.0f) * ((float)H_ * 0.5f);

    int x0 = (int)floorf(x);
    int y0 = (int)floorf(y);
    int x1 = x0 + 1;
    int y1 = y0 + 1;
    x0 = min(max(x0, 0), W_ - 1);
    x1 = min(max(x1, 0), W_ - 1);
    y0 = min(max(y0, 0), H_ - 1);
    y1 = min(max(y1, 0), H_ - 1);

    // Weights computed from the *clipped* coords, exactly as the reference.
    const float x0f = (float)x0, x1f = (float)x1;
    const float y0f = (float)y0, y1f = (float)y1;
    const float wa = (x1f - x) * (y1f - y);
    const float wb = (x1f - x) * (y  - y0f);
    const float wc = (x  - x0f) * (y1f - y);
    const float wd = (x  - x0f) * (y  - y0f);

    // Gather 4 neighbor channel-quads (global_load_b128 each).
    // float4 index of (b, yy, xx, 4q) = (base + yy*W + xx)*8 + q
    const int   base = b * HW_;
    const v4f* im4 = (const v4f*)image;
    const v4f Ia = im4[(base + y0 * W_ + x0) * 8 + q];
    const v4f Ib = im4[(base + y1 * W_ + x0) * 8 + q];
    const v4f Ic = im4[(base + y0 * W_ + x1) * 8 + q];
    const v4f Id = im4[(base + y1 * W_ + x1) * 8 + q];

    v4f o;
    o.x = wa * Ia.x + wb * Ib.x + wc * Ic.x + wd * Id.x;
    o.y = wa * Ia.y + wb * Ib.y + wc * Ic.y + wd * Id.y;
    o.z = wa * Ia.z + wb * Ib.z + wc * Ic.z + wd * Id.z;
    o.w = wa * Ia.w + wb * Ib.w + wc * Ic.w + wd * Id.w;

    // Output: 134 MB write-once, never re-read -> non-temporal store so it
    // doesn't evict the L2-resident image.
    v4f* o4 = (v4f*)out;
    __builtin_nontemporal_store(o, &o4[pix * 8 + q]);
}

extern "C" void kernel_launch(void* const* d_in, const int* in_sizes, int n_in,
                              void* d_out, int out_size, void* d_ws, size_t ws_size,
                              hipStream_t stream) {
    const float* image = (const float*)d_in[0];  // [16,256,256,32] f32
    const float* flow  = (const float*)d_in[1];  // [16,256,256,2]  f32
    float*       out   = (float*)d_out;          // [16,256,256,32] f32

    // B*H*W*8 threads (8 channel-quads per pixel)
    const int total   = B_ * H_ * W_ * 8;        // 8,388,608
    const int block   = 256;                     // 8 wave32 waves
    const int grid    = total / block;           // 32,768 blocks

    bilinear_sampler_kernel<<<grid, block, 0, stream>>>(image, flow, out);
}